// MyModel_87522843560183
// MI455X (gfx1250) — compile-verified
//
#include <hip/hip_runtime.h>

// Problem constants (match reference)
#define VOCAB 300
#define TT    2048
#define BB    4096
#define HID   2

#define BLOCK   64              // threads per block (2 waves), one batch row per thread
#define TC      64              // timesteps staged per chunk
#define NCHUNK  (TT / TC)       // 32 chunks
#define ROWPAD  68              // ints per LDS ids row (64 data + 4 pad -> 272B stride)
#define TAB_ROWS 304            // 300 vocab rows padded to 19 * 16

typedef float v2f __attribute__((ext_vector_type(2)));
typedef float v8f __attribute__((ext_vector_type(8)));

#if __has_builtin(__builtin_amdgcn_tanhf)
// CDNA5 hardware transcendental: V_TANH_F32 (single trans op)
__device__ __forceinline__ float tanh_fast(float x) {
    return __builtin_amdgcn_tanhf(x);
}
__device__ __forceinline__ float sigm(float x) {
    // sigmoid(x) = 0.5*tanh(x/2) + 0.5  -> 1 trans + 2 VALU
    return fmaf(0.5f, __builtin_amdgcn_tanhf(0.5f * x), 0.5f);
}
#else
__device__ __forceinline__ float sigm(float x) {
    // 1/(1+exp(-x)) via v_exp_f32 (exp2) + v_rcp_f32
    float e = __builtin_amdgcn_exp2f(-1.44269504f * x);
    return __builtin_amdgcn_rcpf(1.0f + e);
}
__device__ __forceinline__ float tanh_fast(float x) {
    // tanh(x) = (e-1)/(e+1), e = exp2(2*log2e*x)
    float e = __builtin_amdgcn_exp2f(2.88539008f * x);
    return (e - 1.0f) * __builtin_amdgcn_rcpf(e + 1.0f);
}
#endif

// Issue TC ints (16B x 16) of this lane's row chunk: global -> LDS, async (gfx1250).
__device__ __forceinline__ void stage_chunk(const int* __restrict__ src, unsigned lds_dst) {
#pragma unroll
    for (int j = 0; j < TC / 4; ++j) {
        asm volatile("global_load_async_to_lds_b128 %0, %1, off"
                     :: "v"(lds_dst + (unsigned)(j * 16)), "v"(src + j * 4)
                     : "memory");
    }
}

__device__ __forceinline__ void lstm_step(int id, const float* __restrict__ s_tab,
                                          const float (&u)[16],
                                          float& h0, float& h1, float& c0, float& c1) {
    const float4* tp = (const float4*)(s_tab + id * 8);   // 32B-aligned gate row
    float4 pa = tp[0];   // { i0', i1', f0', f1' }
    float4 pb = tp[1];   // { g0', g1', o0', o1' }
    float zi0 = fmaf(h1, u[8 + 0], fmaf(h0, u[0], pa.x));
    float zi1 = fmaf(h1, u[8 + 1], fmaf(h0, u[1], pa.y));
    float zf0 = fmaf(h1, u[8 + 2], fmaf(h0, u[2], pa.z));
    float zf1 = fmaf(h1, u[8 + 3], fmaf(h0, u[3], pa.w));
    float zg0 = fmaf(h1, u[8 + 4], fmaf(h0, u[4], pb.x));
    float zg1 = fmaf(h1, u[8 + 5], fmaf(h0, u[5], pb.y));
    float zo0 = fmaf(h1, u[8 + 6], fmaf(h0, u[6], pb.z));
    float zo1 = fmaf(h1, u[8 + 7], fmaf(h0, u[7], pb.w));
    float i0 = sigm(zi0), i1 = sigm(zi1);
    float f0 = sigm(zf0), f1 = sigm(zf1);
    float g0 = tanh_fast(zg0), g1 = tanh_fast(zg1);
    float o0 = sigm(zo0), o1 = sigm(zo1);
    c0 = fmaf(f0, c0, i0 * g0);
    c1 = fmaf(f1, c1, i1 * g1);
    h0 = o0 * tanh_fast(c0);
    h1 = o1 * tanh_fast(c1);
}

__global__ __launch_bounds__(BLOCK) void lstm_tinyhid_kernel(
        const int* __restrict__ ids, const float* __restrict__ E,
        const float* __restrict__ W, const float* __restrict__ U,
        const float* __restrict__ bias, float* __restrict__ out) {
    __shared__ float s_tab[TAB_ROWS * 8];       // P[v][g] = E[v,:]@W + b
    __shared__ int   s_ids[2][BLOCK][ROWPAD];   // double-buffered id chunks

    const int tid  = threadIdx.x;
    const int lane = tid & 31;
    const int wave = tid >> 5;

    // ---------------- Build gate-preprojection table with V_WMMA_F32_16X16X4_F32 ----
    {
        const int col = lane & 15;   // N (and M offset selector within lane halves)
        const int khi = lane >> 4;   // 0 -> K={0,1}, 1 -> K={2,3} (zero pad)

        // B fragment: 4x16 (KxN). Rows K0/K1 = W rows; K2/K3 and cols>=8 are zero.
        float bmask = (khi == 0 && col < 8) ? 1.0f : 0.0f;
        int   wcol  = col & 7;
        v2f bfrag;
        bfrag.x = W[wcol] * bmask;          // K=0 row: W[0, n]
        bfrag.y = W[8 + wcol] * bmask;      // K=1 row: W[1, n]

        // C fragment: bias broadcast down rows (all 8 VGPRs identical per lane).
        float cval = (col < 8) ? bias[col] : 0.0f;
        v8f cfrag = {cval, cval, cval, cval, cval, cval, cval, cval};

        const int tstart = (wave == 0) ? 0 : 10;
        const int tend   = (wave == 0) ? 10 : 19;
        for (int tile = tstart; tile < tend; ++tile) {
            int   row   = tile * 16 + col;
            int   rc    = (row < VOCAB) ? row : (VOCAB - 1);
            float amask = (khi == 0 && row < VOCAB) ? 1.0f : 0.0f;
            v2f afrag;
            afrag.x = E[rc * 2 + 0] * amask;   // K=0
            afrag.y = E[rc * 2 + 1] * amask;   // K=1 (khi lanes carry K=2,3 -> zero)
            v8f d = __builtin_amdgcn_wmma_f32_16x16x4_f32(
                        false, afrag, false, bfrag, (short)0, cfrag, false, false);
            if (col < 8) {
                int rbase = tile * 16 + khi * 8;
#pragma unroll
                for (int j = 0; j < 8; ++j)
                    s_tab[(rbase + j) * 8 + col] = d[j];
            }
        }
    }
    __syncthreads();

    // ---------------- Per-lane LSTM over T, ids staged async global->LDS ------------
    const long long row = (long long)blockIdx.x * BLOCK + tid;
    const int*  idrow  = ids + row * TT;
    float*      outrow = out + row * (long long)(TT * HID);

    float u[16];
#pragma unroll
    for (int k = 0; k < 16; ++k) u[k] = U[k];   // uniform -> scalar loads

    stage_chunk(idrow, (unsigned)(uintptr_t)&s_ids[0][tid][0]);

    float h0 = 0.0f, h1 = 0.0f, c0 = 0.0f, c1 = 0.0f;

    for (int c = 0; c < NCHUNK; ++c) {
        if (c + 1 < NCHUNK) {
            stage_chunk(idrow + (c + 1) * TC,
                        (unsigned)(uintptr_t)&s_ids[(c + 1) & 1][tid][0]);
            // 16 new copies in flight; <=16 outstanding means chunk c has landed
            asm volatile("s_wait_asynccnt 16" ::: "memory");
        } else {
            asm volatile("s_wait_asynccnt 0" ::: "memory");
        }

        const int4* ip   = (const int4*)&s_ids[c & 1][tid][0];
        float*      orow = outrow + c * (TC * HID);
#pragma unroll 4
        for (int q = 0; q < TC / 4; ++q) {
            int4 idv = ip[q];
            lstm_step(idv.x, s_tab, u, h0, h1, c0, c1);
            float ha0 = h0, ha1 = h1;
            lstm_step(idv.y, s_tab, u, h0, h1, c0, c1);
            *(float4*)(orow + q * 8)     = make_float4(ha0, ha1, h0, h1);
            lstm_step(idv.z, s_tab, u, h0, h1, c0, c1);
            float hc0 = h0, hc1 = h1;
            lstm_step(idv.w, s_tab, u, h0, h1, c0, c1);
            *(float4*)(orow + q * 8 + 4) = make_float4(hc0, hc1, h0, h1);
        }
    }
}

extern "C" void kernel_launch(void* const* d_in, const int* in_sizes, int n_in,
                              void* d_out, int out_size, void* d_ws, size_t ws_size,
                              hipStream_t stream) {
    const int*   ids = (const int*)d_in[0];
    const float* E   = (const float*)d_in[1];
    const float* W   = (const float*)d_in[2];
    const float* U   = (const float*)d_in[3];
    const float* b   = (const float*)d_in[4];
    float* out = (float*)d_out;
    (void)in_sizes; (void)n_in; (void)out_size; (void)d_ws; (void)ws_size;

    dim3 grid(BB / BLOCK);   // 64 blocks
    dim3 block(BLOCK);       // 64 threads = 2 waves
    hipLaunchKernelGGL(lstm_tinyhid_kernel, grid, block, 0, stream,
                       ids, E, W, U, b, out);
}